// AttentionLayer_29523605192863
// MI455X (gfx1250) — compile-verified
//
#include <hip/hip_runtime.h>
#include <cmath>

// ---------------------------------------------------------------------------
// AttentionLayer for MI455X (gfx1250, wave32) using V_WMMA_F32_16X16X4_F32.
//   proj = tanh(reps @ W) (* diagW for lt)      -> d_ws   [2*B*L, A]
//   out  = postmask(softmax(premask(lt @ rt^T)))-> d_out  [B, L, L]
// ---------------------------------------------------------------------------

typedef __attribute__((ext_vector_type(2))) float v2f;
typedef __attribute__((ext_vector_type(4))) float v4f;
typedef __attribute__((ext_vector_type(8))) float v8f;

#define BB 32
#define LL 1024
#define HH 256
#define AA 128

// ---------------------------------------------------------------------------
// Kernel 1: fused projection GEMM [2*B*L,256]@[256,128] + tanh (+ diagW).
// 256 threads = 8 waves; each wave owns a 16-row x 128-col output tile.
// W staged TRANSPOSED in LDS: WT[a][h], stride 260 floats -> a lane's B frag
// (W[k][col], W[k+1][col]) is one contiguous conflict-free ds_load_b64.
// Per-ct LDS base pointers are hoisted so K offsets fit the DS immediate,
// and fragments are staged 8-at-a-time so DS loads overlap the WMMA chain.
// ---------------------------------------------------------------------------
__global__ __launch_bounds__(256) void proj_tanh_kernel(
    const float* __restrict__ reps_lt, const float* __restrict__ reps_rt,
    const float* __restrict__ W, const float* __restrict__ diagW,
    float* __restrict__ proj)
{
    extern __shared__ float WT[];            // [AA][TSTR] transposed W
    constexpr int TSTR = HH + 4;             // 260 floats; 1040 B (16B aligned)
    const int tid = threadIdx.x;

    // Cooperative transpose copy: coalesced v4f global reads, scalar LDS writes.
    for (int i = tid * 4; i < HH * AA; i += 256 * 4) {
        const int h = i >> 7;                // W is [HH][AA] row-major
        const int a = i & (AA - 1);
        const v4f w = *(const v4f*)(W + h * AA + a);
        WT[(a + 0) * TSTR + h] = w.x;
        WT[(a + 1) * TSTR + h] = w.y;
        WT[(a + 2) * TSTR + h] = w.z;
        WT[(a + 3) * TSTR + h] = w.w;
    }
    __syncthreads();

    const int lane = tid & 31;
    const int wv   = tid >> 5;
    const int lo   = lane & 15;              // row-within-tile / col-within-tile
    const int hi   = lane >> 4;              // K sub-phase selector

    const int row0 = (blockIdx.x * 8 + wv) * 16;   // 16-row tile base (uniform per wave)
    const int grow = row0 + lo;                    // this lane's source row
    const bool is_lt = row0 < BB * LL;             // tile boundary aligned: uniform
    const float* __restrict__ src =
        is_lt ? reps_lt + (size_t)grow * HH
              : reps_rt + (size_t)(grow - BB * LL) * HH;

    // Hoisted per-column-tile LDS base pointers (K offset then fits DS imm).
    const float* bp[8];
#pragma unroll
    for (int ct = 0; ct < 8; ++ct)
        bp[ct] = WT + (ct * 16 + lo) * TSTR + 2 * hi;

    v8f acc[8] = {};                          // 8 column tiles of 16x16 f32

    for (int p = 0; p < HH / 4; ++p) {        // K = 256 in steps of 4
        const int k = 4 * p;
        const v2f a = *(const v2f*)(src + k + 2 * hi);  // A frag (16x4 f32 layout)

        v2f bf[8];                            // stage all 8 B frags: loads overlap
#pragma unroll
        for (int ct = 0; ct < 8; ++ct)
            bf[ct] = *(const v2f*)(bp[ct] + k);
#pragma unroll
        for (int ct = 0; ct < 8; ++ct)
            acc[ct] = __builtin_amdgcn_wmma_f32_16x16x4_f32(
                false, a, false, bf[ct], (short)0, acc[ct], false, false);
    }

    // Epilogue: tanh (+ diag scale for lt rows), coalesced f32 stores.
#pragma unroll
    for (int ct = 0; ct < 8; ++ct) {
        const int col  = ct * 16 + lo;
        const float dw = is_lt ? diagW[col] : 1.0f;
#pragma unroll
        for (int i = 0; i < 8; ++i) {
            const int m = i + 8 * hi;         // C/D layout: VGPR i -> rows i / i+8
            proj[(size_t)(row0 + m) * AA + col] = tanhf(acc[ct][i]) * dw;
        }
    }
}

// ---------------------------------------------------------------------------
// Kernel 2: fused scores (lt @ rt^T) + pre-mask + softmax + post-mask.
// One WG = 16 lt rows x 1024 rt cols of one batch; 8 waves, 128 cols each.
// Score tile lives in LDS (16 x 1028 padded) -> never touches HBM.
// rt slab (512 KB/batch) is re-read by 64 WGs but stays resident in 192MB L2.
// ---------------------------------------------------------------------------
__global__ __launch_bounds__(256) void attn_softmax_kernel(
    const float* __restrict__ proj, const float* __restrict__ mask_lt,
    const float* __restrict__ mask_rt, float* __restrict__ out)
{
    extern __shared__ float S[];              // [16][SSTR]
    constexpr int SSTR = LL + 4;              // pad: rows i and i+8 land 32 banks apart
    __shared__ float red[16][17];
    __shared__ float mlt_s[16];

    const int tid = threadIdx.x;
    const int b   = blockIdx.y;
    const int lt0 = blockIdx.x * 16;

    if (tid < 16) mlt_s[tid] = mask_lt[(size_t)b * LL + lt0 + tid];
    __syncthreads();

    const int lane = tid & 31;
    const int wv   = tid >> 5;
    const int lo   = lane & 15;
    const int hi   = lane >> 4;

    const float* __restrict__ ltp = proj + ((size_t)b * LL + lt0 + lo) * AA;
    const float* __restrict__ rtb = proj + ((size_t)(BB * LL) + (size_t)b * LL) * AA;
    const float* __restrict__ mrt = mask_rt + (size_t)b * LL;

    // Resident A fragments: 16x128 lt tile = 32 x float2 per lane (64 VGPRs).
    v2f afrag[32];
#pragma unroll
    for (int p = 0; p < 32; ++p)
        afrag[p] = *(const v2f*)(ltp + 4 * p + 2 * hi);

    // ---- Score phase: this wave covers cols [wv*128, wv*128+128) ----
    for (int ct = 0; ct < 8; ++ct) {
        const int cbase = wv * 128 + ct * 16;
        const float* __restrict__ rtp = rtb + (size_t)(cbase + lo) * AA + 2 * hi;
        v8f c = {};
#pragma unroll
        for (int q = 0; q < 4; ++q) {         // K = 128 in 4 chunks of 8 steps
            v2f bf[8];                        // stage 8 frags: loads stay in flight
#pragma unroll
            for (int u = 0; u < 8; ++u)
                bf[u] = *(const v2f*)(rtp + 4 * (8 * q + u));
#pragma unroll
            for (int u = 0; u < 8; ++u)
                c = __builtin_amdgcn_wmma_f32_16x16x4_f32(
                    false, afrag[8 * q + u], false, bf[u], (short)0, c, false, false);
        }
        const float mr = mrt[cbase + lo];
#pragma unroll
        for (int i = 0; i < 8; ++i) {
            const int m = i + 8 * hi;
            S[m * SSTR + cbase + lo] = c[i] * mlt_s[m] * mr;   // pre-softmax mask
        }
    }
    __syncthreads();

    // ---- Softmax phase: 16 threads per row, float4 strided over 1024 cols ----
    const int r   = tid >> 4;
    const int c16 = tid & 15;

    float mx = -3.402823466e38f;
#pragma unroll 4
    for (int j = 0; j < 16; ++j) {
        const v4f v = *(const v4f*)(S + r * SSTR + c16 * 4 + 64 * j);
        mx = fmaxf(mx, fmaxf(fmaxf(v.x, v.y), fmaxf(v.z, v.w)));
    }
    red[r][c16] = mx;
    __syncthreads();
    float rmax = red[r][0];
    for (int t = 1; t < 16; ++t) rmax = fmaxf(rmax, red[r][t]);
    __syncthreads();

    float sum = 0.f;
    for (int j = 0; j < 16; ++j) {
        float* p4 = S + r * SSTR + c16 * 4 + 64 * j;
        v4f v = *(const v4f*)p4;
        v.x = expf(v.x - rmax);
        v.y = expf(v.y - rmax);
        v.z = expf(v.z - rmax);
        v.w = expf(v.w - rmax);
        sum += (v.x + v.y) + (v.z + v.w);
        *(v4f*)p4 = v;                        // each thread owns its region: no race
    }
    red[r][c16] = sum;
    __syncthreads();
    float rsum = 0.f;
    for (int t = 0; t < 16; ++t) rsum += red[r][t];

    const float mlr = mlt_s[r] / rsum;        // 1/sum folded with post-softmax lt mask
    float* __restrict__ orow = out + ((size_t)b * LL + lt0 + r) * (size_t)LL;
    for (int j = 0; j < 16; ++j) {
        const int cc = c16 * 4 + 64 * j;
        const v4f e  = *(const v4f*)(S + r * SSTR + cc);
        const v4f m4 = *(const v4f*)(mrt + cc);
        v4f o;
        o.x = e.x * mlr * m4.x;
        o.y = e.y * mlr * m4.y;
        o.z = e.z * mlr * m4.z;
        o.w = e.w * mlr * m4.w;
        __builtin_nontemporal_store(o, (v4f*)(orow + cc));  // write-once 128 MB: NT
    }
}

// ---------------------------------------------------------------------------
extern "C" void kernel_launch(void* const* d_in, const int* in_sizes, int n_in,
                              void* d_out, int out_size, void* d_ws, size_t ws_size,
                              hipStream_t stream)
{
    (void)in_sizes; (void)n_in; (void)out_size; (void)ws_size;

    const float* reps_lt = (const float*)d_in[0];   // [B,L,H]
    const float* reps_rt = (const float*)d_in[1];   // [B,L,H]
    const float* mask_lt = (const float*)d_in[2];   // [B,L]
    const float* mask_rt = (const float*)d_in[3];   // [B,L]
    const float* attn_k  = (const float*)d_in[4];   // [H,A]
    const float* diagW   = (const float*)d_in[5];   // [1,1,A]
    float* outp = (float*)d_out;                    // [B,L,L]
    float* proj = (float*)d_ws;                     // [2*B*L, A] f32 = 32 MB scratch

    // Projection: 2*B*L/16 = 4096 row-tiles, 8 per WG -> 512 WGs.
    const size_t ldsW = (size_t)AA * (HH + 4) * sizeof(float);   // ~130 KB
    proj_tanh_kernel<<<dim3(512), 256, ldsW, stream>>>(
        reps_lt, reps_rt, attn_k, diagW, proj);

    // Fused attention+softmax: (L/16) x B = 64 x 32 WGs.
    const size_t ldsS = (size_t)16 * (LL + 4) * sizeof(float);   // ~64 KB
    attn_softmax_kernel<<<dim3(LL / 16, BB), 256, ldsS, stream>>>(
        proj, mask_lt, mask_rt, outp);
}